// FraudGNN_48481590837453
// MI455X (gfx1250) — compile-verified
//
#include <hip/hip_runtime.h>
#include <hip/hip_bf16.h>

typedef __attribute__((ext_vector_type(2))) float v2f;
typedef __attribute__((ext_vector_type(8))) float v8f;

// ---------------------------------------------------------------------------
// Zero fill (grid-stride)
// ---------------------------------------------------------------------------
__global__ void zero_kernel(float* __restrict__ p, long long n) {
    long long i = (long long)blockIdx.x * blockDim.x + threadIdx.x;
    long long stride = (long long)gridDim.x * blockDim.x;
    for (; i < n; i += stride) p[i] = 0.0f;
}

// ---------------------------------------------------------------------------
// Degree: deg[dst] += 1 per edge
// ---------------------------------------------------------------------------
__global__ void degree_kernel(const int* __restrict__ ei, float* __restrict__ deg, int E) {
    int e = blockIdx.x * blockDim.x + threadIdx.x;
    if (e < E) atomicAdd(&deg[ei[E + e]], 1.0f);
}

// ---------------------------------------------------------------------------
// Scatter-add of source features into agg[dst]. F = feature width (mult of 4).
// One thread handles one (edge, 4-float chunk).
// ---------------------------------------------------------------------------
template <int F>
__global__ void scatter_kernel(const int* __restrict__ ei, const float* __restrict__ feat,
                               float* __restrict__ agg, int E) {
    long long t = (long long)blockIdx.x * blockDim.x + threadIdx.x;
    int e = (int)(t / (F / 4));
    int f4 = (int)(t % (F / 4)) * 4;
    if (e >= E) return;
    int src = ei[e];
    int dst = ei[E + e];
    const float4 v = *(const float4*)(feat + (size_t)src * F + f4);
    float* p = agg + (size_t)dst * F + f4;
    atomicAdd(p + 0, v.x);
    atomicAdd(p + 1, v.y);
    atomicAdd(p + 2, v.z);
    atomicAdd(p + 3, v.w);
}

// ---------------------------------------------------------------------------
// SAGE combine with f32 WMMA:
//   out = relu( (agg/deg) @ Wl + bl + xin @ Wr )
// Block handles 16 nodes (M tile). NOUT/16 waves; each wave owns one 16-col
// output tile and iterates K in steps of 4 using V_WMMA_F32_16X16X4_F32.
//
// A layout (16x4 f32): lane L holds row M=L%16; VGPR0/1 = K pair selected by
// L/16 (lanes 0-15: K=0,1; lanes 16-31: K=2,3).
// B layout (4x16 f32): lane L holds col N=L%16, same K-pair split.
// C/D (16x16 f32): lane L col N=L%16, VGPR v = row v + 8*(L/16).
// ---------------------------------------------------------------------------
template <int K, int NOUT>
__global__ __launch_bounds__(32 * (NOUT / 16)) void sage_combine_kernel(
    const float* __restrict__ xin, const float* __restrict__ agg,
    const float* __restrict__ deg, const float* __restrict__ Wl,
    const float* __restrict__ bl, const float* __restrict__ Wr,
    float* __restrict__ out, int Nn) {
    __shared__ float s_inv[16];
    __shared__ float s_a[16 * K];  // mean-aggregated features
    __shared__ float s_x[16 * K];  // raw input features

    const int node_base = blockIdx.x * 16;
    const int tid = threadIdx.x;
    const int lane = tid & 31;
    const int wave = tid >> 5;            // N tile index
    const int m = lane & 15;              // A-row / node within tile
    const int khalf = lane >> 4;          // selects K pair (0 -> k,k+1; 1 -> k+2,k+3)
    const int ncol = wave * 16 + m;       // output column (lane%16)

    if (tid < 16) {
        int nd = node_base + tid;
        if (nd > Nn - 1) nd = Nn - 1;
        s_inv[tid] = 1.0f / fmaxf(deg[nd], 1.0f);
    }
    __syncthreads();

    const int nthreads = 32 * (NOUT / 16);
    for (int i = tid; i < 16 * K; i += nthreads) {
        int nd = i / K;
        int f = i - nd * K;
        int gnode = node_base + nd;
        if (gnode > Nn - 1) gnode = Nn - 1;
        s_a[i] = agg[(size_t)gnode * K + f] * s_inv[nd];
        s_x[i] = xin[(size_t)gnode * K + f];
    }
    __syncthreads();

    v8f acc = {};
#pragma unroll
    for (int k = 0; k < K; k += 4) {
        const int kb = k + khalf * 2;
        v2f a, b;
        // agg @ Wl
        a.x = s_a[m * K + kb];
        a.y = s_a[m * K + kb + 1];
        b.x = Wl[(size_t)kb * NOUT + ncol];
        b.y = Wl[(size_t)(kb + 1) * NOUT + ncol];
        acc = __builtin_amdgcn_wmma_f32_16x16x4_f32(false, a, false, b,
                                                    (short)0, acc, false, false);
        // x @ Wr
        a.x = s_x[m * K + kb];
        a.y = s_x[m * K + kb + 1];
        b.x = Wr[(size_t)kb * NOUT + ncol];
        b.y = Wr[(size_t)(kb + 1) * NOUT + ncol];
        acc = __builtin_amdgcn_wmma_f32_16x16x4_f32(false, a, false, b,
                                                    (short)0, acc, false, false);
    }

    const float bias = bl[ncol];
#pragma unroll
    for (int v = 0; v < 8; ++v) {
        int mrow = v + khalf * 8;
        int gnode = node_base + mrow;
        if (gnode < Nn) {
            float val = acc[v] + bias;
            out[(size_t)gnode * NOUT + ncol] = fmaxf(val, 0.0f);
        }
    }
}

// ---------------------------------------------------------------------------
// Final FC (64 -> 1) + sigmoid
// ---------------------------------------------------------------------------
__global__ void fc_sigmoid_kernel(const float* __restrict__ h2,
                                  const float* __restrict__ Wfc,
                                  const float* __restrict__ bfc,
                                  float* __restrict__ out, int Nn) {
    int i = blockIdx.x * blockDim.x + threadIdx.x;
    if (i >= Nn) return;
    float s = bfc[0];
    const float* hp = h2 + (size_t)i * 64;
#pragma unroll
    for (int k = 0; k < 64; ++k) s += hp[k] * Wfc[k];
    out[i] = 1.0f / (1.0f + __expf(-s));
}

// ---------------------------------------------------------------------------
// Launcher
// ---------------------------------------------------------------------------
extern "C" void kernel_launch(void* const* d_in, const int* in_sizes, int n_in,
                              void* d_out, int out_size, void* d_ws, size_t ws_size,
                              hipStream_t stream) {
    const float* x    = (const float*)d_in[0];
    const int*   ei   = (const int*)d_in[1];
    const float* W1l  = (const float*)d_in[2];
    const float* b1   = (const float*)d_in[3];
    const float* W1r  = (const float*)d_in[4];
    const float* W2l  = (const float*)d_in[5];
    const float* b2   = (const float*)d_in[6];
    const float* W2r  = (const float*)d_in[7];
    const float* Wfc  = (const float*)d_in[8];
    const float* bfc  = (const float*)d_in[9];
    float* outp = (float*)d_out;

    const int Nn = in_sizes[0] / 64;  // 100000
    const int Ee = in_sizes[1] / 2;   // 1600000

    // Workspace layout (floats): deg[N] | agg[N*128] | h1[N*128] | h2[N*64]
    float* deg = (float*)d_ws;
    float* agg = deg + Nn;
    float* h1  = agg + (size_t)Nn * 128;
    float* h2  = h1 + (size_t)Nn * 128;

    const int ZB = 2048, TB = 256;

    // ---- Layer 1 ----
    zero_kernel<<<ZB, TB, 0, stream>>>(deg, (long long)Nn * 129);  // deg + agg(128 lanes worth)
    degree_kernel<<<(Ee + TB - 1) / TB, TB, 0, stream>>>(ei, deg, Ee);
    {
        long long t = (long long)Ee * (64 / 4);
        scatter_kernel<64><<<(int)((t + TB - 1) / TB), TB, 0, stream>>>(ei, x, agg, Ee);
    }
    sage_combine_kernel<64, 128><<<(Nn + 15) / 16, 256, 0, stream>>>(
        x, agg, deg, W1l, b1, W1r, h1, Nn);

    // ---- Layer 2 ----
    zero_kernel<<<ZB, TB, 0, stream>>>(agg, (long long)Nn * 128);
    {
        long long t = (long long)Ee * (128 / 4);
        scatter_kernel<128><<<(int)((t + TB - 1) / TB), TB, 0, stream>>>(ei, h1, agg, Ee);
    }
    sage_combine_kernel<128, 64><<<(Nn + 15) / 16, 128, 0, stream>>>(
        h1, agg, deg, W2l, b2, W2r, h2, Nn);

    // ---- Head ----
    fc_sigmoid_kernel<<<(Nn + TB - 1) / TB, TB, 0, stream>>>(h2, Wfc, bfc, outp, Nn);
}